// BeatPulseTransportCore_13829794693465
// MI455X (gfx1250) — compile-verified
//
#include <hip/hip_runtime.h>
#include <math.h>

// BeatPulseTransportCore — fused inject/advect/diffuse/fade over [B,160,3] f32.
// Memory-bound (~252 MB traffic -> ~10.8us floor at 23.3 TB/s). Uses CDNA5
// async global->LDS copies (ASYNCcnt) to stage tiles, computes from LDS,
// stores with b128.

static constexpr int RBINS   = 160;
static constexpr int TILE_B  = 16;
static constexpr int NTHR    = 256;
static constexpr int ROWF    = RBINS * 3;          // 480 floats per row
static constexpr int CHUNKS_PER_ROW = ROWF * 4 / 16; // 120 x 16B chunks

__global__ __launch_bounds__(NTHR)
void beatpulse_kernel(const float* __restrict__ history,
                      const float* __restrict__ color,
                      const float* __restrict__ p_offset,
                      const float* __restrict__ p_persist,
                      const float* __restrict__ p_diff,
                      const float* __restrict__ p_dt,
                      const float* __restrict__ p_amount,
                      const float* __restrict__ p_spread,
                      float* __restrict__ out,
                      int B)
{
    __shared__ __align__(16) float tile[TILE_B * ROWF];

    const int tid = threadIdx.x;
    const int b0  = blockIdx.x * TILE_B;
    int rows = B - b0;
    if (rows <= 0) return;
    if (rows > TILE_B) rows = TILE_B;

    // ---- scalar parameters (replicates reference f32 math) ----
    float dt        = fminf(fmaxf(p_dt[0], 0.0f), 0.05f);
    float dt_scale  = dt * 60.0f;
    float dt_offset = p_offset[0] * dt_scale;
    float dt_persist = powf(p_persist[0], dt_scale);
    float amount = fminf(fmaxf(p_amount[0], 0.0f), 1.0f);
    float spread = fminf(fmaxf(p_spread[0], 0.0f), 1.0f);
    float tight  = 1.0f - spread;
    float wk0 = 0.5f + 0.4f * tight;
    float wk1 = 0.2f * spread + 0.05f;
    float wk2 = 0.12f * spread;
    float wk3 = 0.06f * spread;
    float wk4 = 0.02f * spread;
    float kdiff = 0.15f * p_diff[0];
    float cdiff = 1.0f - 2.0f * kdiff;
    int   K = (int)floorf(dt_offset);

    // ---- stage 1: async bulk copy of tile rows into LDS (CDNA5 path) ----
    {
        const unsigned nchunk = (unsigned)(rows * CHUNKS_PER_ROW);
        unsigned long long gbase =
            (unsigned long long)(const void*)(history + (size_t)b0 * ROWF);
        // low 32 bits of a generic pointer to __shared__ == LDS byte offset
        unsigned lbase = (unsigned)(unsigned long long)(const void*)&tile[0];
        for (unsigned i = tid; i < nchunk; i += NTHR) {
            unsigned voff = i * 16u;
            unsigned ldsa = lbase + voff;
            asm volatile("global_load_async_to_lds_b128 %0, %1, %2 offset:0"
                         :: "v"(ldsa), "v"(voff), "s"(gbase)
                         : "memory");
        }
        asm volatile("s_wait_asynccnt 0" ::: "memory");
    }
    __syncthreads();

    // ---- stage 2: inject energy into bins 0..4 (clamp happens at read) ----
    if (tid < rows * 15) {
        int row = tid / 15;
        int rem = tid % 15;
        int bin = rem / 3;
        int c   = rem % 3;
        float wv = (bin == 0) ? wk0 : (bin == 1) ? wk1 : (bin == 2) ? wk2
                 : (bin == 3) ? wk3 : wk4;
        float energy = color[(size_t)(b0 + row) * 3 + c] * amount;
        tile[row * ROWF + bin * 3 + c] += energy * wv;
    }
    __syncthreads();

    // ---- stage 3: fused advect + diffuse + fade, 4 bins x 3ch per task ----
    const int ntask = rows * (RBINS / 4);
    for (int t = tid; t < ntask; t += NTHR) {
        int row = t / (RBINS / 4);
        int j0  = (t % (RBINS / 4)) * 4;
        int sbase = j0 - K - 2;   // 8 consecutive source bins cover all taps

        // per-source advection weights (channel independent), exactly as ref:
        // new_pos = s + dt_offset; left = clip(floor,0,R-2); frac = np - left
        float wl[8], wr[8];
        #pragma unroll
        for (int m = 0; m < 8; ++m) {
            int s = sbase + m;
            float wlv = 0.0f, wrv = 0.0f;
            if (s >= 0 && s < RBINS) {
                float np    = (float)s + dt_offset;
                bool  valid = (np >= 0.0f) && (np < (float)(RBINS - 1));
                int lefti = (int)floorf(np);
                if (lefti < 0) lefti = 0;
                if (lefti > RBINS - 2) lefti = RBINS - 2;
                float frac = np - (float)lefti;
                // guard: this source must land on its expected bin j0-2+m
                if (valid && lefti == (j0 - 2 + m)) {
                    wlv = (1.0f - frac) * dt_persist;
                    wrv = frac * dt_persist;
                }
            }
            wl[m] = wlv;
            wr[m] = wrv;
        }

        const float* tp = &tile[row * ROWF];
        float4 o[3];
        float* ov = (float*)o;    // ov[q*3+c] = out(j0+q, c)

        #pragma unroll
        for (int c = 0; c < 3; ++c) {
            float hcv[8];
            #pragma unroll
            for (int m = 0; m < 8; ++m) {
                int s = sbase + m;
                float v = 0.0f;
                if (s >= 0 && s < RBINS) {
                    v = tp[s * 3 + c];
                    v = fminf(fmaxf(v, 0.0f), 1.0f);   // post-inject clamp
                }
                hcv[m] = v;
            }
            // A(p) for p = j0-1 .. j0+4 ; zero outside [0,R) (diffusion pad)
            float Ap[6];
            #pragma unroll
            for (int d = 0; d < 6; ++d) {
                int p = j0 - 1 + d;
                float a = hcv[d + 1] * wl[d + 1] + hcv[d] * wr[d];
                Ap[d] = (p >= 0 && p < RBINS) ? a : 0.0f;
            }
            #pragma unroll
            for (int q = 0; q < 4; ++q) {
                int j = j0 + q;
                float val = cdiff * Ap[q + 1] + kdiff * Ap[q] + kdiff * Ap[q + 2];
                if (j >= RBINS - 8) {
                    float tt = (float)(RBINS - 1 - j) * 0.125f;
                    val *= tt * tt;
                }
                ov[q * 3 + c] = val;
            }
        }

        float4* op = (float4*)(out + ((size_t)(b0 + row) * RBINS + j0) * 3);
        op[0] = o[0];
        op[1] = o[1];
        op[2] = o[2];
    }
}

extern "C" void kernel_launch(void* const* d_in, const int* in_sizes, int n_in,
                              void* d_out, int out_size, void* d_ws, size_t ws_size,
                              hipStream_t stream)
{
    const float* history = (const float*)d_in[0];
    const float* color   = (const float*)d_in[1];
    const float* off     = (const float*)d_in[2];
    const float* pers    = (const float*)d_in[3];
    const float* diff    = (const float*)d_in[4];
    const float* dt      = (const float*)d_in[5];
    const float* amount  = (const float*)d_in[6];
    const float* spread  = (const float*)d_in[7];
    float* out = (float*)d_out;

    int B = in_sizes[0] / ROWF;
    int grid = (B + TILE_B - 1) / TILE_B;
    hipLaunchKernelGGL(beatpulse_kernel, dim3(grid), dim3(NTHR), 0, stream,
                       history, color, off, pers, diff, dt, amount, spread,
                       out, B);
}